// multigraph_undirected_sep_63651415327268
// MI455X (gfx1250) — compile-verified
//
#include <hip/hip_runtime.h>
#include <cstddef>
#include <cstdint>

typedef __attribute__((ext_vector_type(2))) float v2f;
typedef __attribute__((ext_vector_type(8))) float v8f;

#define ALPHA_F 3.0f
#define TOPK    20
#define NTOT    8192
#define DIM     64

#define BM   128
#define BN   256
#define KT   16
#define LDST 20   // padded LDS row stride (floats): 80B = 16B-aligned, bank-conflict-free

// ---------------------------------------------------------------------------
// nodevec: out[r,d] = tanh(ALPHA * (sum_k emb[r,k]*lin_w[d,k] + lin_b[d]))
// ---------------------------------------------------------------------------
__global__ __launch_bounds__(256) void nv_kernel(const float* __restrict__ emb,
                                                 const float* __restrict__ lw,
                                                 const float* __restrict__ lb,
                                                 float* __restrict__ out) {
  __shared__ float s_lw[DIM * 65];   // stride 65 -> conflict-free column reads
  __shared__ float s_emb[4][DIM];
  __shared__ float s_b[DIM];
  const int tid = threadIdx.x;
  for (int i = tid; i < DIM * DIM; i += 256) {
    int d = i >> 6, k = i & 63;
    s_lw[d * 65 + k] = lw[i];
  }
  if (tid < DIM) s_b[tid] = lb[tid];
  const int row0 = blockIdx.x * 4;
  for (int i = tid; i < 4 * DIM; i += 256) {
    int r = i >> 6, k = i & 63;
    s_emb[r][k] = emb[(size_t)(row0 + r) * DIM + k];
  }
  __syncthreads();
  const int d  = tid & 63;
  const int ry = tid >> 6;
  float sum = s_b[d];
  const float* e = s_emb[ry];
  const float* w = &s_lw[d * 65];
#pragma unroll
  for (int k = 0; k < DIM; ++k) sum = fmaf(e[k], w[k], sum);
  out[(size_t)(row0 + ry) * DIM + d] = tanhf(ALPHA_F * sum);
}

// ---------------------------------------------------------------------------
// async global -> LDS staging (CDNA5 GLOBAL_LOAD_ASYNC_TO_LDS_B128, ASYNCcnt)
// ---------------------------------------------------------------------------
template <int ROWS>
__device__ __forceinline__ void async_tile(const float* __restrict__ g, int ldg,
                                           int rowBase, int kcol,
                                           float* __restrict__ sbuf, int tid) {
  // ROWS x KT floats; 256 threads x (ROWS*KT/1024) x b128 each
#pragma unroll
  for (int i = 0; i < (ROWS * KT) / 1024; ++i) {
    int idx = tid + i * 256;
    int r = idx >> 2;                 // 0..ROWS-1
    int c = (idx & 3) << 2;           // 0,4,8,12
    const float* gp = g + (size_t)(rowBase + r) * ldg + (kcol + c);
    unsigned lds = (unsigned)(uintptr_t)(sbuf + r * LDST + c);  // low32 = LDS offset
    asm volatile("global_load_async_to_lds_b128 %0, %1, off"
                 :: "v"(lds), "v"(gp)
                 : "memory");
  }
}

__device__ __forceinline__ void wait_async_all() {
  asm volatile("s_wait_asynccnt 0x0" ::: "memory");
}

// ---------------------------------------------------------------------------
// WMMA compute: 64x64 per wave per KT-tile (V_WMMA_F32_16X16X4_F32)
// ---------------------------------------------------------------------------
__device__ __forceinline__ void mma_tile(const float* __restrict__ As,
                                         const float* __restrict__ Bs,
                                         int wrow, int wcol, int lm, int lk,
                                         v8f acc[4][4]) {
#pragma unroll
  for (int kk = 0; kk < KT; kk += 4) {
    v2f a[4], b[4];
#pragma unroll
    for (int mi = 0; mi < 4; ++mi)
      a[mi] = *(const v2f*)(As + (wrow + mi * 16 + lm) * LDST + kk + lk);
#pragma unroll
    for (int ni = 0; ni < 4; ++ni)
      b[ni] = *(const v2f*)(Bs + (wcol + ni * 16 + lm) * LDST + kk + lk);
#pragma unroll
    for (int mi = 0; mi < 4; ++mi)
#pragma unroll
      for (int ni = 0; ni < 4; ++ni)
        acc[mi][ni] = __builtin_amdgcn_wmma_f32_16x16x4_f32(
            false, a[mi], false, b[ni], (short)0, acc[mi][ni], false, false);
  }
}

// virtual-K source select: [0,Kp) -> prior operands, [Kp,Kp+64) -> nodevecs
__device__ __forceinline__ void tile_src(int k0, int Kp,
                                         const float* priA, const float* priB,
                                         const float* nv1, const float* nv2,
                                         const float** A, const float** B,
                                         int* ld, int* kc) {
  if (k0 < Kp) { *A = priA; *B = priB; *ld = Kp;  *kc = k0; }
  else         { *A = nv1;  *B = nv2;  *ld = DIM; *kc = k0 - Kp; }
}

// out[row_off+m, col_off+n] = relu(tanh(ALPHA*( nv1[m,:]·nv2[n,:]
//                           + [Kp>0] (priA[m,:]·priB[n,:] + bias[n]) )))
__global__ __launch_bounds__(256) void adj_kernel(
    const float* __restrict__ nv1, const float* __restrict__ nv2,
    const float* __restrict__ priA, const float* __restrict__ priB,
    const float* __restrict__ bias, float* __restrict__ out,
    int Kp, int row_off, int col_off, int has_prior) {
  __shared__ float As[2][BM * LDST];   // 2 x 10240 B
  __shared__ float Bs[2][BN * LDST];   // 2 x 20480 B   (61440 B total)

  const int tid  = threadIdx.x;
  const int wid  = tid >> 5;
  const int lane = tid & 31;
  const int lm   = lane & 15;          // fragment row (M or N)
  const int lk   = (lane >> 4) << 1;   // fragment K pair: 0 or 2

  const int tileM = blockIdx.y * BM;
  const int tileN = blockIdx.x * BN;
  const int wrow  = (wid >> 2) * 64;   // 0 or 64
  const int wcol  = (wid & 3) * 64;    // 0,64,128,192

  v8f acc[4][4];
  const v8f vzero = {0.f, 0.f, 0.f, 0.f, 0.f, 0.f, 0.f, 0.f};
#pragma unroll
  for (int mi = 0; mi < 4; ++mi)
#pragma unroll
    for (int ni = 0; ni < 4; ++ni) acc[mi][ni] = vzero;

  const int Ktot = Kp + DIM;           // Kp and 64 are both multiples of KT
  const int nT   = Ktot / KT;

  // prologue: stage tile 0 into buffer 0
  {
    const float *A, *B; int ld, kc;
    tile_src(0, Kp, priA, priB, nv1, nv2, &A, &B, &ld, &kc);
    async_tile<BM>(A, ld, tileM, kc, As[0], tid);
    async_tile<BN>(B, ld, tileN, kc, Bs[0], tid);
  }
  wait_async_all();
  __syncthreads();

  for (int t = 0; t < nT; ++t) {
    if (t + 1 < nT) {                  // wave-uniform; EXEC stays all-ones
      const float *A, *B; int ld, kc;
      tile_src((t + 1) * KT, Kp, priA, priB, nv1, nv2, &A, &B, &ld, &kc);
      async_tile<BM>(A, ld, tileM, kc, As[(t + 1) & 1], tid);
      async_tile<BN>(B, ld, tileN, kc, Bs[(t + 1) & 1], tid);
    }
    mma_tile(As[t & 1], Bs[t & 1], wrow, wcol, lm, lk, acc);
    wait_async_all();                  // own tile t+1 writes landed
    __syncthreads();                   // everyone's landed; reads of buf[t&1] done
  }

  const int l16 = lane >> 4;
#pragma unroll
  for (int mi = 0; mi < 4; ++mi) {
#pragma unroll
    for (int ni = 0; ni < 4; ++ni) {
      const int colL = tileN + wcol + ni * 16 + lm;
      const float bv = has_prior ? bias[colL] : 0.f;
      const size_t colG = (size_t)(col_off + colL);
#pragma unroll
      for (int v = 0; v < 8; ++v) {
        const int rowL = tileM + wrow + mi * 16 + (v + 8 * l16);
        float x = acc[mi][ni][v] + bv;
        x = tanhf(ALPHA_F * x);
        x = x > 0.f ? x : 0.f;
        out[(size_t)(row_off + rowL) * NTOT + colG] = x;
      }
    }
  }
}

// ---------------------------------------------------------------------------
// exact per-row top-K mask (stable, lowest-index-first ties) in-place
// ---------------------------------------------------------------------------
__global__ __launch_bounds__(256) void topk_kernel(float* __restrict__ out) {
  __shared__ float srow[NTOT];         // 32 KB
  __shared__ int   s_pre[256];
  __shared__ int   s_cnt;
  const int tid  = threadIdx.x;
  const int lane = tid & 31;
  float* rp = out + (size_t)blockIdx.x * NTOT;

  for (int i = tid; i < NTOT / 4; i += 256)
    *(float4*)&srow[i * 4] = *(const float4*)&rp[i * 4];
  __syncthreads();

  const int  chunk = NTOT / 256;       // 32 contiguous elems per thread
  float* my = &srow[tid * chunk];

  // Binary search on bit pattern for the K-th largest value.
  // All values are >= +0 after relu, so uint order == float order.
  unsigned t = 0u;
  for (int bit = 30; bit >= 0; --bit) {
    unsigned cand = t | (1u << bit);
    if (cand > 0x3F800000u) continue;  // values are <= 1.0
    const float cv = __uint_as_float(cand);
    if (tid == 0) s_cnt = 0;
    __syncthreads();
    int c = 0;
    for (int j = 0; j < chunk; ++j) c += (my[j] >= cv);
    for (int off = 16; off; off >>= 1) c += __shfl_xor(c, off);
    if (lane == 0) atomicAdd(&s_cnt, c);
    __syncthreads();
    const int total = s_cnt;
    __syncthreads();
    if (total >= TOPK) t = cand;
  }
  const float vk = __uint_as_float(t);

  // strictly-greater count -> how many ties to keep
  if (tid == 0) s_cnt = 0;
  __syncthreads();
  {
    int c = 0;
    for (int j = 0; j < chunk; ++j) c += (my[j] > vk);
    for (int off = 16; off; off >>= 1) c += __shfl_xor(c, off);
    if (lane == 0) atomicAdd(&s_cnt, c);
  }
  __syncthreads();
  const int keep = TOPK - s_cnt;
  __syncthreads();

  // index-ordered rank of ties: per-chunk counts + exclusive prefix
  int ce = 0;
  for (int j = 0; j < chunk; ++j) ce += (my[j] == vk);
  s_pre[tid] = ce;
  __syncthreads();
  if (tid == 0) {
    int run = 0;
    for (int i = 0; i < 256; ++i) { int v = s_pre[i]; s_pre[i] = run; run += v; }
  }
  __syncthreads();
  int rank = s_pre[tid];
  for (int j = 0; j < chunk; ++j) {
    float x = my[j];
    float o;
    if (x > vk)       o = x;
    else if (x == vk) { o = (rank < keep) ? x : 0.f; rank++; }
    else              o = 0.f;
    my[j] = o;
  }
  __syncthreads();
  for (int i = tid; i < NTOT / 4; i += 256)
    *(float4*)&rp[i * 4] = *(const float4*)&srow[i * 4];
}

// ---------------------------------------------------------------------------
extern "C" void kernel_launch(void* const* d_in, const int* in_sizes, int n_in,
                              void* d_out, int out_size, void* d_ws, size_t ws_size,
                              hipStream_t stream) {
  (void)in_sizes; (void)n_in; (void)out_size; (void)ws_size;
  const float* emb0 = (const float*)d_in[0];
  const float* lw0  = (const float*)d_in[1];
  const float* lb0  = (const float*)d_in[2];
  const float* emb1 = (const float*)d_in[3];
  const float* lw1  = (const float*)d_in[4];
  const float* lb1  = (const float*)d_in[5];
  const float* emb2 = (const float*)d_in[6];
  const float* lw2  = (const float*)d_in[7];
  const float* lb2  = (const float*)d_in[8];
  const float* emb3 = (const float*)d_in[9];
  const float* lw3  = (const float*)d_in[10];
  const float* lb3  = (const float*)d_in[11];
  const float* Ww0  = (const float*)d_in[12];
  const float* Wb0  = (const float*)d_in[13];
  const float* pre0 = (const float*)d_in[14];
  const float* Ww1  = (const float*)d_in[15];
  const float* Wb1  = (const float*)d_in[16];
  const float* pre1 = (const float*)d_in[17];
  // d_in[18] = idx (identity arange, not needed)
  float* out = (float*)d_out;

  const int LA = 5120, LB = 3072;
  float* ws    = (float*)d_ws;
  float* nvA0  = ws;                        // [5120,64]  block(0,0) both sides
  float* nvB1a = nvA0  + (size_t)LA * DIM;  // [5120,64]  block(0,1) rows
  float* nvB1b = nvB1a + (size_t)LA * DIM;  // [3072,64]  block(0,1) cols
  float* nvC2a = nvB1b + (size_t)LB * DIM;  // [3072,64]  block(1,0) rows
  float* nvC2b = nvC2a + (size_t)LB * DIM;  // [5120,64]  block(1,0) cols
  float* nvD3  = nvC2b + (size_t)LA * DIM;  // [3072,64]  block(1,1) both sides

  nv_kernel<<<LA / 4, 256, 0, stream>>>(emb0, lw0, lb0, nvA0);
  nv_kernel<<<LA / 4, 256, 0, stream>>>(emb1, lw1, lb1, nvB1a);
  nv_kernel<<<LB / 4, 256, 0, stream>>>(emb2, lw1, lb1, nvB1b);
  nv_kernel<<<LB / 4, 256, 0, stream>>>(emb2, lw2, lb2, nvC2a);
  nv_kernel<<<LA / 4, 256, 0, stream>>>(emb1, lw2, lb2, nvC2b);
  nv_kernel<<<LB / 4, 256, 0, stream>>>(emb3, lw3, lb3, nvD3);

  adj_kernel<<<dim3(LA / BN, LA / BM), 256, 0, stream>>>(
      nvA0, nvA0, pre0, Ww0, Wb0, out, LA, 0, 0, 1);
  adj_kernel<<<dim3(LB / BN, LA / BM), 256, 0, stream>>>(
      nvB1a, nvB1b, nullptr, nullptr, nullptr, out, 0, 0, LA, 0);
  adj_kernel<<<dim3(LA / BN, LB / BM), 256, 0, stream>>>(
      nvC2a, nvC2b, nullptr, nullptr, nullptr, out, 0, LA, 0, 0);
  adj_kernel<<<dim3(LB / BN, LB / BM), 256, 0, stream>>>(
      nvD3, nvD3, pre1, Ww1, Wb1, out, LB, LA, LA, 1);

  topk_kernel<<<NTOT, 256, 0, stream>>>(out);
}